// ChunkAggregator_1125281431613
// MI455X (gfx1250) — compile-verified
//
#include <hip/hip_runtime.h>
#include <hip/hip_bf16.h>
#include <stdint.h>

// ---------------- problem constants (from reference) ----------------
#define VOCAB   32000
#define BLK     128
#define NH      2
#define NB      8
#define NL      16384
#define NBLK    (NL / BLK)            // 128 blocks per batch row
#define NSEG    (NB * NBLK)           // 1024 segments
#define CATROW  (NBLK * NH)           // 256
#define NEWROW  (CATROW + NL * NH)    // 33024
#define LDSV    32768                 // padded LDS histogram (pow2 -> clean loops)
#define THREADS 512

// ---------------- CDNA5 feature probes (reported via #warning) ----------------
#if __has_builtin(__builtin_amdgcn_tensor_store_from_lds)
#  define HAVE_TDM 1
#else
#  define HAVE_TDM 0
#endif
#if __has_builtin(__builtin_amdgcn_global_store_async_from_lds_b128)
#  define HAVE_ASYNC 1
#else
#  define HAVE_ASYNC 0
#endif
#if __has_builtin(__builtin_amdgcn_s_wait_tensorcnt)
#  define HAVE_TENSORCNT 1
#else
#  define HAVE_TENSORCNT 0
#endif
#if __has_builtin(__builtin_amdgcn_s_wait_asynccnt)
#  define HAVE_ASYNCCNT 1
#else
#  define HAVE_ASYNCCNT 0
#endif
#if __has_include(<hip/amd_detail/amd_gfx1250_TDM.h>)
#  define TDM_6ARG 1
#else
#  define TDM_6ARG 0
#endif

// --- compile-only telemetry: these show in hipcc stderr, build still succeeds ---
#if HAVE_TDM
#  if TDM_6ARG
#    warning "CDNA5-PROBE: writeout path = TDM tensor_store_from_lds (6-arg, therock headers)"
#  else
#    warning "CDNA5-PROBE: writeout path = TDM tensor_store_from_lds (5-arg, ROCm 7.2)"
#  endif
#elif HAVE_ASYNC
#  warning "CDNA5-PROBE: writeout path = global_store_async_from_lds_b128"
#else
#  warning "CDNA5-PROBE: writeout path = PLAIN STORES (no TDM/async builtin found)"
#endif
#if !HAVE_TENSORCNT
#  warning "CDNA5-PROBE: s_wait_tensorcnt builtin MISSING"
#endif
#if !HAVE_ASYNCCNT
#  warning "CDNA5-PROBE: s_wait_asynccnt builtin MISSING"
#endif

typedef __attribute__((ext_vector_type(4))) unsigned int u32x4;
typedef __attribute__((ext_vector_type(4))) int          i32x4;
typedef __attribute__((ext_vector_type(8))) int          i32x8;

// =====================================================================
// Kernel 1: per-segment histogram. 1024 workgroups x 512 threads (16 waves).
//   LDS histogram 32768 f32 (128 KiB; 2 WGs resident per 320 KiB WGP LDS)
//   zero: 16 unrolled ds_store_b128 per thread (no bounds checks)
//   scatter: 256 x ds_add_f32 (non-returning LDS float atomic)
//   writeout: one TDM D# per workgroup -> 128,000 B LDS->HBM DMA
// =====================================================================
__global__ __launch_bounds__(THREADS)
void hist_kernel(const int* __restrict__ tokens, float* __restrict__ hist_out)
{
    __shared__ __align__(16) float lds_hist[LDSV];
    const int seg = blockIdx.x;        // 0..1023  (= b*128 + block)
    const int tid = threadIdx.x;

    // ---- zero LDS: exactly 8192 float4 = 16 per thread, fully unrolled ----
    float4* lds4 = (float4*)lds_hist;
    const float4 z = make_float4(0.f, 0.f, 0.f, 0.f);
#pragma unroll
    for (int i = 0; i < LDSV / 4 / THREADS; ++i)   // 16 iterations
        lds4[tid + i * THREADS] = z;
    __syncthreads();

    // ---- scatter: the 256 tokens of this segment are contiguous ----
    if (tid < BLK * NH) {
        int tok = tokens[(size_t)seg * (BLK * NH) + tid];
        if ((unsigned)tok < (unsigned)VOCAB)
            atomicAdd(&lds_hist[tok], 1.0f);       // -> ds_add_f32 (no rtn)
    }
    __syncthreads();

    float* dst = hist_out + (size_t)seg * VOCAB;

#if HAVE_TDM
    // ---- TDM DMA: one descriptor stores the 32000 valid floats ----
    if (tid < 32) {   // wave 0 only, full EXEC
        uint32_t lds_off = (uint32_t)(uintptr_t)&lds_hist[0]; // generic low 32b = LDS byte offset
        uint64_t ga      = (uint64_t)(uintptr_t)dst;

        // D# group 0: count=1 | lds_addr | global_addr[56:0] | type=2
        u32x4 g0;
        g0[0] = 1u;                                   // count=1, is_restore=0
        g0[1] = lds_off;                              // lds_addr (bytes)
        g0[2] = (uint32_t)(ga & 0xFFFFFFFFu);         // global_addr[31:0]
        g0[3] = (uint32_t)((ga >> 32) & 0x01FFFFFFu)  // global_addr[56:32]
              | (2u << 30);                           // type = 2 ("image")

        // D# group 1: data_size=4B; tensor_dim0=tile_dim0=32000;
        // tensor_dim1=tile_dim1=1; strides=32000.
        i32x8 g1;
        g1[0] = (int)(2u << 16);                      // data_size=2 (4B), wg_mask=0
        g1[1] = (int)((VOCAB & 0xFFFF) << 16);        // tensor_dim0[15:0] @ b63:48
        g1[2] = (int)(1u << 16);                      // dim0[31:16]=0, tensor_dim1[15:0]=1
        g1[3] = (int)((VOCAB & 0xFFFF) << 16);        // dim1[31:16]=0, tile_dim0=32000
        g1[4] = 1;                                    // tile_dim1=1, tile_dim2=0
        g1[5] = VOCAB;                                // tensor_dim0_stride[31:0]
        g1[6] = (int)((VOCAB & 0xFFFF) << 16);        // stride0[47:32]=0, stride1[15:0]=32000
        g1[7] = 0;                                    // stride1[47:16]=0

        i32x4 gz = {0, 0, 0, 0};                      // groups 2/3 unused (2D tensor)
#if TDM_6ARG
        i32x8 g4 = {0, 0, 0, 0, 0, 0, 0, 0};
        __builtin_amdgcn_tensor_store_from_lds(g0, g1, gz, gz, g4, 0);
#else
        __builtin_amdgcn_tensor_store_from_lds(g0, g1, gz, gz, 0);
#endif
#if HAVE_TENSORCNT
        __builtin_amdgcn_s_wait_tensorcnt(0);
#endif
    }
#elif HAVE_ASYNC
    // ---- async LDS->global B128 stores (ASYNCcnt), no VGPR staging ----
    for (int i = tid; i < VOCAB / 4; i += THREADS) {
        __builtin_amdgcn_global_store_async_from_lds_b128(
            (__attribute__((address_space(1))) void*)(dst + 4 * (size_t)i),
            (__attribute__((address_space(3))) void*)&lds4[i], 0, 0);
    }
#if HAVE_ASYNCCNT
    __builtin_amdgcn_s_wait_asynccnt(0);
#endif
#else
    // ---- fallback: packed LDS read + global_store_b128 ----
    float4* dst4 = (float4*)dst;
    for (int i = tid; i < VOCAB / 4; i += THREADS)
        dst4[i] = lds4[i];
#endif
}

// =====================================================================
// Kernel 2: new_tokens + cat_ids (pure copy/gather, ~2 MB traffic).
// =====================================================================
__global__ __launch_bounds__(256)
void tok_kernel(const int* __restrict__ tokens,
                float* __restrict__ d_new, float* __restrict__ d_cat)
{
    int idx = blockIdx.x * 256 + threadIdx.x;
    if (idx >= NB * NEWROW) return;
    int b = idx / NEWROW;
    int j = idx - b * NEWROW;
    int v;
    if (j < CATROW) {
        int nb = j >> 1, h = j & 1;                    // token at (b, nb*128, h)
        v = tokens[(size_t)b * (NL * NH) + (size_t)nb * (BLK * NH) + h];
        d_cat[b * CATROW + j] = (float)v;
    } else {
        v = tokens[(size_t)b * (NL * NH) + (j - CATROW)];
    }
    d_new[idx] = (float)v;
}

// =====================================================================
extern "C" void kernel_launch(void* const* d_in, const int* in_sizes, int n_in,
                              void* d_out, int out_size, void* d_ws, size_t ws_size,
                              hipStream_t stream)
{
    const int* tokens = (const int*)d_in[0];   // (8,16384,2) int32
    // d_in[1] (cat_embed_f) unused by reference; d_in[2] only supplies VOCAB.
    float* out    = (float*)d_out;
    float* d_new  = out;                                   // 8*33024
    float* d_cat  = out + (size_t)NB * NEWROW;             // 8*256
    float* d_hist = d_cat + (size_t)NB * CATROW;           // 1024*32000

    hist_kernel<<<NSEG, THREADS, 0, stream>>>(tokens, d_hist);
    tok_kernel<<<(NB * NEWROW + 255) / 256, 256, 0, stream>>>(tokens, d_new, d_cat);
}